// SumAggregation_29575144800264
// MI455X (gfx1250) — compile-verified
//
#include <hip/hip_runtime.h>

// Scatter-sum (segment_sum): out[index[e], :] += x[e, :]
//   x:     (625000, 128) f32
//   index: (625000,)     i32
//   out:   (50000, 128)  f32
//
// Memory-bound: ~320MB streamed reads of x, atomics resolved in L2
// (25.6MB output working set << 192MB L2). Wave32 mapping: one wave
// per edge row, lane l owns floats [4l, 4l+4) of the 128-wide row.
//
// blockDim=32 (single wave per workgroup): trivial prologue so the disasm
// snippet reaches the loop body; no barrier resource allocated; up to
// 32 single-wave workgroups resident per WGP.

typedef float v4f __attribute__((ext_vector_type(4)));

#define D_FEAT 128

__global__ __launch_bounds__(32) void scatter_sum_kernel(
    const v4f* __restrict__ x4,        // x viewed as (N_EDGES, 32) float4
    const int* __restrict__ index,     // (N_EDGES,)
    float* __restrict__ out,           // (N_NODES, 128)
    int n_edges)
{
    const int lane    = threadIdx.x;   // 0..31
    const int wstride = gridDim.x;

    for (int e = blockIdx.x; e < n_edges; e += wstride) {
        // Prefetch next iteration's row chunk (gfx1250: global_prefetch_b8).
        int en = e + wstride;
        if (en < n_edges) {
            __builtin_prefetch(&x4[(size_t)en * 32 + lane], 0, 0);
        }

        // Wave-uniform segment id (all 32 lanes hit the same dword; one L2 req).
        int idx = index[e];

        // Streamed once -> non-temporal b128 load, don't pollute L2 where the
        // accumulator lines live.
        v4f v = __builtin_nontemporal_load(&x4[(size_t)e * 32 + lane]);

        // 512B contiguous atomic region per wave; hardware f32 add at L2.
        float* o = out + (size_t)idx * D_FEAT + lane * 4;
        unsafeAtomicAdd(o + 0, v.x);
        unsafeAtomicAdd(o + 1, v.y);
        unsafeAtomicAdd(o + 2, v.z);
        unsafeAtomicAdd(o + 3, v.w);
    }
}

__global__ __launch_bounds__(256) void zero_out_kernel(v4f* __restrict__ out4, int n4) {
    int i = blockIdx.x * blockDim.x + threadIdx.x;
    if (i < n4) {
        v4f z = {0.0f, 0.0f, 0.0f, 0.0f};
        out4[i] = z;
    }
}

extern "C" void kernel_launch(void* const* d_in, const int* in_sizes, int n_in,
                              void* d_out, int out_size, void* d_ws, size_t ws_size,
                              hipStream_t stream) {
    const float* x     = (const float*)d_in[0];
    const int*   index = (const int*)d_in[1];
    // d_in[2] is dim_size (scalar); out_size already encodes N_NODES * 128.
    float* out = (float*)d_out;

    const int n_edges = in_sizes[0] / D_FEAT;

    // 1) Zero the accumulator (harness poisons d_out; must re-init every call).
    const int n4 = out_size / 4;
    const int zblocks = (n4 + 255) / 256;
    zero_out_kernel<<<zblocks, 256, 0, stream>>>((v4f*)out, n4);

    // 2) Scatter-add. 32768 single-wave workgroups, ~19 edge rows each:
    //    enough parallelism to saturate HBM, few enough for prefetch reuse.
    scatter_sum_kernel<<<32768, 32, 0, stream>>>(
        (const v4f*)x, index, out, n_edges);
}